// HungarianMatcher_25984552141569
// MI455X (gfx1250) — compile-verified
//
#include <hip/hip_runtime.h>
#include <cstdint>

// ---------------------------------------------------------------------------
// DETR HungarianMatcher cost matrix on MI455X (gfx1250, wave32).
//   cost[b,q,t] = -1*softmax(logits)[b,q,label[b,t]]
//               + 5*L1(pred_box[b,q], tgt_box[b,t])        (cxcywh)
//               - 2*GIoU(pred_box[b,q], tgt_box[b,t])      (xyxy)
//
// v3 changes vs round 2:
//  * One-hot B-fragment build normalized: labm = lab - khalf computed once
//    per t-tile, compares go against literal constants kk*4 / kk*4+1
//    -> 4 VALU per WMMA step instead of 6 (no v_or index materialization).
//  * Everything else unchanged: TDM tensor_load_to_lds staging of the logits
//    tile, 8-threads/row parallel softmax with 1/sum folded into A-fragments,
//    V_WMMA_F32_16X16X4_F32 one-hot GEMM for the class-prob gather, hoisted
//    per-lane q-box VGPR arrays, coalesced t-contiguous stores.
// ---------------------------------------------------------------------------

constexpr int Qd = 300, Td = 200, Cd = 80;
constexpr int QT = 16;                      // query rows per block
constexpr int NTT = (Td + 15) / 16;         // 13 t-tiles of 16
constexpr int BLOCK = 128;                  // 4 wave32

typedef float v2f __attribute__((ext_vector_type(2)));
typedef float v8f __attribute__((ext_vector_type(8)));
typedef unsigned int v4u __attribute__((ext_vector_type(4)));
typedef int v4i __attribute__((ext_vector_type(4)));
typedef int v8i __attribute__((ext_vector_type(8)));

#if defined(__HIP_DEVICE_COMPILE__) && __has_builtin(__builtin_amdgcn_tensor_load_to_lds) && __has_builtin(__builtin_amdgcn_s_wait_tensorcnt)
#define HAVE_TDM 1
#else
#define HAVE_TDM 0
#endif

#if defined(__HIP_DEVICE_COMPILE__) && __has_builtin(__builtin_amdgcn_wmma_f32_16x16x4_f32)
#define HAVE_WMMA4 1
#else
#define HAVE_WMMA4 0
#endif

__global__ __launch_bounds__(BLOCK)
void hungarian_cost_kernel(const float* __restrict__ logits,
                           const float* __restrict__ pboxes,
                           const int*   __restrict__ tlabels,
                           const float* __restrict__ tboxes,
                           float* __restrict__ out)
{
    __shared__ float raw[QT * Cd];              // raw logits tile (TDM target)
    __shared__ float probs[QT * (Cd + 1)];      // UNSCALED exp rows, stride 81
    __shared__ float red_max[BLOCK];            // softmax partial maxes
    __shared__ float red_sum[BLOCK];            // softmax partial sums
    __shared__ float q_rs[QT];                  // per-row 1/sum(exp)
    __shared__ float t_cx[Td], t_cy[Td], t_w[Td], t_h[Td];
    __shared__ float t_x0[Td], t_y0[Td], t_x1[Td], t_y1[Td], t_ar[Td];
    __shared__ int   t_lab[Td];
    __shared__ float q_cx[QT], q_cy[QT], q_w[QT], q_h[QT];
    __shared__ float q_x0[QT], q_y0[QT], q_x1[QT], q_y1[QT], q_ar[QT];

    const int b      = blockIdx.y;
    const int q0     = blockIdx.x * QT;
    const int nrowsQ = min(QT, Qd - q0);
    const int tid    = threadIdx.x;
    const int lane   = tid & 31;
    const int wave   = tid >> 5;

    const size_t logit_base = ((size_t)b * Qd + q0) * (size_t)Cd;

#if HAVE_TDM
    // ---- issue async TDM load of the 16x80 f32 logits tile (wave 0 only) ----
    if (wave == 0) {
        const uint64_t ga = (uint64_t)(uintptr_t)(logits + logit_base);
        const uint32_t la = (uint32_t)(uintptr_t)(&raw[0]);   // LDS byte offset
        v4u d0;
        d0[0] = 1u;                                           // count=1, is_restore=0, gather off
        d0[1] = la;                                           // lds_addr
        d0[2] = (uint32_t)ga;                                 // global_addr[31:0]
        d0[3] = ((uint32_t)(ga >> 32) & 0x01FFFFFFu) | 0x80000000u; // addr[56:32] | type=2
        v8i d1;
        d1[0] = (2 << 16);                 // workgroup_mask=0, data_size=2 (4 bytes)
        d1[1] = (Cd & 0xffff) << 16;       // tensor_dim0 = 80 (lo16 @ bits 63:48)
        d1[2] = (nrowsQ & 0xffff) << 16;   // tensor_dim0 hi=0 | tensor_dim1 lo16
        d1[3] = (Cd & 0xffff) << 16;       // tensor_dim1 hi=0 | tile_dim0 = 80
        d1[4] = (nrowsQ & 0xffff);         // tile_dim1 = nrowsQ, tile_dim2 = 0
        d1[5] = Cd;                        // tensor_dim0_stride = 80 (lo32)
        d1[6] = 0;
        d1[7] = 0;
        v4i d2 = {0, 0, 0, 0};
        v4i d3 = {0, 0, 0, 0};
#if __clang_major__ >= 23
        v8i d4 = {0, 0, 0, 0, 0, 0, 0, 0};
        __builtin_amdgcn_tensor_load_to_lds(d0, d1, d2, d3, d4, 0);
#else
        __builtin_amdgcn_tensor_load_to_lds(d0, d1, d2, d3, 0);
#endif
    }
#else
    for (int i = tid; i < nrowsQ * Cd; i += BLOCK)
        raw[i] = logits[logit_base + i];
#endif

    // ---- stage per-target precomputes (overlaps with TDM transfer) ----
    for (int t = tid; t < Td; t += BLOCK) {
        t_lab[t] = tlabels[(size_t)b * Td + t];
        const float4 bb = reinterpret_cast<const float4*>(tboxes)[(size_t)b * Td + t];
        const float cx = bb.x, cy = bb.y, w = bb.z, h = bb.w;
        t_cx[t] = cx; t_cy[t] = cy; t_w[t] = w; t_h[t] = h;
        t_x0[t] = cx - 0.5f * w; t_y0[t] = cy - 0.5f * h;
        t_x1[t] = cx + 0.5f * w; t_y1[t] = cy + 0.5f * h;
        t_ar[t] = fmaxf(w, 0.f) * fmaxf(h, 0.f);
    }
    // ---- stage per-query precomputes ----
    if (tid < nrowsQ) {
        const float4 bb = reinterpret_cast<const float4*>(pboxes)[(size_t)b * Qd + q0 + tid];
        const float cx = bb.x, cy = bb.y, w = bb.z, h = bb.w;
        q_cx[tid] = cx; q_cy[tid] = cy; q_w[tid] = w; q_h[tid] = h;
        q_x0[tid] = cx - 0.5f * w; q_y0[tid] = cy - 0.5f * h;
        q_x1[tid] = cx + 0.5f * w; q_y1[tid] = cy + 0.5f * h;
        q_ar[tid] = fmaxf(w, 0.f) * fmaxf(h, 0.f);
    }

#if HAVE_TDM
    __builtin_amdgcn_s_wait_tensorcnt(0);
#endif
    __syncthreads();

    // ---- parallel softmax: 8 threads per row, all 128 threads busy ----
    const int m8 = tid >> 3;                    // row 0..15
    const int j8 = tid & 7;                     // 0..7 within row
    const bool rowValid = (m8 < nrowsQ);

    float pmax = -3.402823466e38f;
    if (rowValid)
        for (int c = j8; c < Cd; c += 8) pmax = fmaxf(pmax, raw[m8 * Cd + c]);
    red_max[tid] = pmax;
    __syncthreads();

    float psum = 0.f;
    if (rowValid) {
        float mx = red_max[m8 * 8 + 0];
#pragma unroll
        for (int k = 1; k < 8; ++k) mx = fmaxf(mx, red_max[m8 * 8 + k]);
        for (int c = j8; c < Cd; c += 8) {
            const float e = __expf(raw[m8 * Cd + c] - mx);
            probs[m8 * (Cd + 1) + c] = e;       // unscaled; 1/sum folded at preload
            psum += e;
        }
    } else {
        for (int c = j8; c < Cd; c += 8) probs[m8 * (Cd + 1) + c] = 0.f;
    }
    red_sum[tid] = psum;
    __syncthreads();

    if (j8 == 0) {
        float s = 0.f;
        if (rowValid) {
#pragma unroll
            for (int k = 0; k < 8; ++k) s += red_sum[m8 * 8 + k];
            q_rs[m8] = 1.0f / s;
        } else {
            q_rs[m8] = 0.f;
        }
    }
    __syncthreads();

    const int n     = lane & 15;          // N index (t within tile) == M index for A rows
    const int khalf = (lane >> 4) << 1;   // lanes 0-15 -> K {0,1}; lanes 16-31 -> K {2,3}
    const int qhi   = (lane >> 4) << 3;   // C/D: lanes 16-31 hold M = r+8

#if HAVE_WMMA4
    // A-fragments for the whole q-tile, with 1/sum folded in here.
    const float rs_n = q_rs[n];
    v2f afrag[20];
#pragma unroll
    for (int kk = 0; kk < 20; ++kk) {
        const int k = kk * 4 + khalf;
        afrag[kk][0] = probs[n * (Cd + 1) + k]     * rs_n;
        afrag[kk][1] = probs[n * (Cd + 1) + k + 1] * rs_n;
    }
#endif

    // ---- hoist per-lane q-box values into VGPRs (loop-invariant over t) ----
    float hcx[8], hcy[8], hw[8], hh[8], hx0[8], hy0[8], hx1[8], hy1[8], har[8];
#pragma unroll
    for (int r = 0; r < 8; ++r) {
        const int ql = r + qhi;           // always < QT: LDS read safe; store guarded later
        hcx[r] = q_cx[ql]; hcy[r] = q_cy[ql]; hw[r] = q_w[ql]; hh[r] = q_h[ql];
        hx0[r] = q_x0[ql]; hy0[r] = q_y0[ql]; hx1[r] = q_x1[ql]; hy1[r] = q_y1[ql];
        har[r] = q_ar[ql];
    }

    // ---- main loop: each wave takes t-tiles round-robin ----
    for (int tt = wave; tt < NTT; tt += BLOCK / 32) {
        const int  t0     = tt * 16;
        const int  tIdx   = t0 + n;
        const bool tvalid = tIdx < Td;
        const int  tc     = tvalid ? tIdx : (Td - 1);     // clamped -> loads always safe
        int lab = t_lab[tc];                              // unconditional ds_load
        if (!tvalid) lab = -1;                            // one-hot column all zero

#if HAVE_WMMA4
        // Normalize once: compare against literal constants kk*4 / kk*4+1
        // (labm < 0 for invalid lanes never matches any constant).
        const int labm = lab - khalf;
        v8f acc = {0.f, 0.f, 0.f, 0.f, 0.f, 0.f, 0.f, 0.f};
#pragma unroll
        for (int kk = 0; kk < 20; ++kk) {
            v2f bf;                                       // B one-hot fragment
            bf[0] = (labm == kk * 4)     ? 1.0f : 0.0f;
            bf[1] = (labm == kk * 4 + 1) ? 1.0f : 0.0f;
            acc = __builtin_amdgcn_wmma_f32_16x16x4_f32(
                false, afrag[kk], false, bf, (short)0, acc, false, false);
        }
#endif

        const float tcx = t_cx[tc], tcy = t_cy[tc], tw = t_w[tc], th = t_h[tc];
        const float tx0 = t_x0[tc], ty0 = t_y0[tc], tx1 = t_x1[tc], ty1 = t_y1[tc];
        const float tar = t_ar[tc];

#pragma unroll
        for (int r = 0; r < 8; ++r) {
            const int ql = r + qhi;
            // compute unconditionally (garbage rows never stored), guard only the store
            const float l1 = fabsf(hcx[r] - tcx) + fabsf(hcy[r] - tcy)
                           + fabsf(hw[r]  - tw ) + fabsf(hh[r]  - th );
            const float iw    = fmaxf(fminf(hx1[r], tx1) - fmaxf(hx0[r], tx0), 0.f);
            const float ih    = fmaxf(fminf(hy1[r], ty1) - fmaxf(hy0[r], ty0), 0.f);
            const float inter = iw * ih;
            const float uni   = har[r] + tar - inter;
            const float iou   = inter * __builtin_amdgcn_rcpf(uni);
            const float ew    = fmaxf(fmaxf(hx1[r], tx1) - fminf(hx0[r], tx0), 0.f);
            const float eh    = fmaxf(fmaxf(hy1[r], ty1) - fminf(hy0[r], ty0), 0.f);
            const float enc   = ew * eh;
            const float giou  = iou - (enc - uni) * __builtin_amdgcn_rcpf(enc);
#if HAVE_WMMA4
            const float prob = acc[r];
#else
            const float prob = probs[ql * (Cd + 1) + max(lab, 0)] * q_rs[ql];
#endif
            const float cost = -1.0f * prob + 5.0f * l1 - 2.0f * giou;
            if (tvalid && ql < nrowsQ)
                out[((size_t)b * Qd + (q0 + ql)) * (size_t)Td + tIdx] = cost;
        }
    }
}

extern "C" void kernel_launch(void* const* d_in, const int* in_sizes, int n_in,
                              void* d_out, int out_size, void* d_ws, size_t ws_size,
                              hipStream_t stream) {
    (void)n_in; (void)out_size; (void)d_ws; (void)ws_size;
    const float* logits  = (const float*)d_in[0];
    const float* pboxes  = (const float*)d_in[1];
    const int*   tlabels = (const int*)d_in[2];
    const float* tboxes  = (const float*)d_in[3];
    float* out = (float*)d_out;

    const int Bn = in_sizes[0] / (Qd * Cd);     // 512
    dim3 grid((Qd + QT - 1) / QT, Bn);          // 19 x 512
    hungarian_cost_kernel<<<grid, BLOCK, 0, stream>>>(logits, pboxes, tlabels, tboxes, out);
}